// DWT_1D_3882650435603
// MI455X (gfx1250) — compile-verified
//
#include <hip/hip_runtime.h>

// Haar DWT (N=32, C=64, L1=8192):
//   lfc[k] =  s*x[2k] + s*x[2k+1]
//   hfc[k] = -s*x[2k] + s*x[2k+1]      with s = 1/sqrt(2)
// The reference's dense (4096x8192) einsum matrices are 2-banded, so the op
// collapses to a 2-tap filter: 128 MiB of traffic, ~33 MFLOP -> pure
// bandwidth problem (~5.5us at 23.3 TB/s). We run the filter through
// V_WMMA_F32_16X16X4_F32 (matrix pipe is otherwise idle):
//   D(16x16) = A(16x4) * B(4x16) + 0, B[k][n] = x[chunk + 4n + k],
//   A rows: r0=[s,s,0,0] r1=[0,0,s,s] r2=[-s,s,0,0] r3=[0,0,-s,s], rest 0.
// Software-pipelined: 4 B-loads issued back-to-back, then 4 WMMAs, then one
// predicated store block (EXEC is all-1s at every WMMA).

typedef __attribute__((ext_vector_type(2))) float v2f;
typedef __attribute__((ext_vector_type(8))) float v8f;

__launch_bounds__(256)
__global__ void haar_dwt1d_wmma(const float* __restrict__ x,
                                float* __restrict__ lfc,
                                float* __restrict__ hfc)
{
    const int  lane      = threadIdx.x & 31;      // wave32
    const int  waveInBlk = threadIdx.x >> 5;      // 8 waves / 256-thread block
    const long long waveGlobal = (long long)blockIdx.x * 8 + waveInBlk;
    const long long inBase     = waveGlobal * 256;   // 4 chunks x 64 floats

    const float s = 0.70710678118654752440f;

    // A (16x4 f32), ISA layout: lane n (0-15) holds A[n][0..1],
    // lane 16+n holds A[n][2..3]. Only 4 lanes carry nonzeros.
    v2f a; a[0] = 0.0f; a[1] = 0.0f;
    if (lane == 0 || lane == 17) { a[0] =  s; a[1] = s; }
    if (lane == 2 || lane == 19) { a[0] = -s; a[1] = s; }

    // B (4x16 f32) layout: V0 = K0(lanes0-15)/K2(lanes16-31), V1 = K1/K3
    // => lane L loads contiguous float2 at chunk-offset 4*(L&15) + 2*(L>>4).
    const int n  = lane & 15;
    const int hi = lane >> 4;

    const float* bp = x + inBase + (4 * n + 2 * hi);

    // Issue all 4 loads before any WMMA (4 b64 loads in flight per wave).
    const v2f b0 = *(const v2f*)(bp);
    const v2f b1 = *(const v2f*)(bp + 64);
    const v2f b2 = *(const v2f*)(bp + 128);
    const v2f b3 = *(const v2f*)(bp + 192);

    v8f c = {};
    const v8f d0 = __builtin_amdgcn_wmma_f32_16x16x4_f32(false, a, false, b0,
                                                         (short)0, c, false, false);
    const v8f d1 = __builtin_amdgcn_wmma_f32_16x16x4_f32(false, a, false, b1,
                                                         (short)0, c, false, false);
    const v8f d2 = __builtin_amdgcn_wmma_f32_16x16x4_f32(false, a, false, b2,
                                                         (short)0, c, false, false);
    const v8f d3 = __builtin_amdgcn_wmma_f32_16x16x4_f32(false, a, false, b3,
                                                         (short)0, c, false, false);

    // D layout: VGPR r = row r (lanes 0-15) / row r+8 (lanes 16-31).
    // Lane n (<16), chunk j: d[0..3] = {lfc[2n],lfc[2n+1],hfc[2n],hfc[2n+1]}
    // at output index inBase/2 + 32*j + 2n. Lanes 16-31 hold zero rows.
    if (lane < 16) {
        const long long outK = (inBase >> 1) + 2 * n;
        float* lp = lfc + outK;
        float* hp = hfc + outK;
        v2f t;
        t[0] = d0[0]; t[1] = d0[1]; *(v2f*)(lp)      = t;
        t[0] = d1[0]; t[1] = d1[1]; *(v2f*)(lp + 32) = t;
        t[0] = d2[0]; t[1] = d2[1]; *(v2f*)(lp + 64) = t;
        t[0] = d3[0]; t[1] = d3[1]; *(v2f*)(lp + 96) = t;
        t[0] = d0[2]; t[1] = d0[3]; *(v2f*)(hp)      = t;
        t[0] = d1[2]; t[1] = d1[3]; *(v2f*)(hp + 32) = t;
        t[0] = d2[2]; t[1] = d2[3]; *(v2f*)(hp + 64) = t;
        t[0] = d3[2]; t[1] = d3[3]; *(v2f*)(hp + 96) = t;
    }
}

extern "C" void kernel_launch(void* const* d_in, const int* in_sizes, int n_in,
                              void* d_out, int out_size, void* d_ws, size_t ws_size,
                              hipStream_t stream)
{
    (void)n_in; (void)out_size; (void)d_ws; (void)ws_size;

    const float* x = (const float*)d_in[0];
    // d_in[1] / d_in[2] (dense Haar matrices) are ignored: band structure is
    // known analytically.

    const long long total = (long long)in_sizes[0];   // 32*64*8192 = 16,777,216
    float* lfc = (float*)d_out;                       // first tuple element
    float* hfc = (float*)d_out + total / 2;           // second tuple element

    // Each wave handles 256 input floats; 8 waves per 256-thread block.
    const long long waves  = total / 256;             // 65,536 (exact)
    const long long blocks = waves / 8;               // 8,192  (exact)

    haar_dwt1d_wmma<<<dim3((unsigned)blocks), dim3(256), 0, stream>>>(x, lfc, hfc);
}